// GIN_60198261621206
// MI455X (gfx1250) — compile-verified
//
#include <hip/hip_runtime.h>
#include <hip/hip_bf16.h>
#include <math.h>

typedef __attribute__((ext_vector_type(2))) float v2f;
typedef __attribute__((ext_vector_type(8))) float v8f;

#define HID_DIM 128
#define LSTR    132        // LDS row stride (floats): conflict-free & 16B-aligned
#define BN_EPS  1e-5f

__device__ __forceinline__ v8f wmma_f32_k4(v2f a, v2f b, v8f c) {
  // D = A(16x4 f32) * B(4x16 f32) + C(16x16 f32), full fp32 precision
  return __builtin_amdgcn_wmma_f32_16x16x4_f32(
      /*neg_a=*/false, a, /*neg_b=*/false, b,
      /*c_mod=*/(short)0, c, /*reuse_a=*/false, /*reuse_b=*/false);
}

// ---------------------------------------------------------------------------
// Edge scatter: agg[dst] += x[src].  32 threads per edge, float4 per thread.
// x/agg are 25.6MB each -> fully L2-resident on MI455X (192MB L2).
// ---------------------------------------------------------------------------
__global__ __launch_bounds__(256)
void scatter_add_kernel(const float* __restrict__ x,
                        const int* __restrict__ src,
                        const int* __restrict__ dst,
                        float* __restrict__ agg, int nEdges) {
  int tid = blockIdx.x * 256 + threadIdx.x;
  int e = tid >> 5;
  int q = tid & 31;
  if (e < nEdges) {
    int s = src[e];
    int d = dst[e];
    const float4 v = *(const float4*)(x + (size_t)s * HID_DIM + q * 4);
    float* o = agg + (size_t)d * HID_DIM + q * 4;
    atomicAdd(o + 0, v.x);
    atomicAdd(o + 1, v.y);
    atomicAdd(o + 2, v.z);
    atomicAdd(o + 3, v.w);
  }
}

// ---------------------------------------------------------------------------
// Fused GIN MLP layer: out = BN(relu(relu(((1+eps)x+agg)@W1+b1)@W2+b2))
// Block = 128 threads (4 waves), 16 node rows per block.
// Each wave computes 2 column tiles (16x16) per GEMM with f32 WMMA.
// `out` may alias `agg` (rows are read into LDS before being overwritten).
// ---------------------------------------------------------------------------
__global__ __launch_bounds__(128)
void gin_mlp_kernel(const float* __restrict__ x, const float* __restrict__ agg,
                    const float* __restrict__ epsArr, int layer,
                    const float* __restrict__ W1_, const float* __restrict__ b1_,
                    const float* __restrict__ W2_, const float* __restrict__ b2_,
                    const float* __restrict__ g_,  const float* __restrict__ be_,
                    const float* __restrict__ m_,  const float* __restrict__ v_,
                    float* __restrict__ out, int nNodes) {
  __shared__ float h0[16 * LSTR];
  __shared__ float h1[16 * LSTR];

  const int tid     = threadIdx.x;
  const int rowBase = blockIdx.x * 16;

  const float eps1 = 1.0f + epsArr[layer];
  const float* W1 = W1_ + (size_t)layer * HID_DIM * HID_DIM;
  const float* b1 = b1_ + layer * HID_DIM;
  const float* W2 = W2_ + (size_t)layer * HID_DIM * HID_DIM;
  const float* b2 = b2_ + layer * HID_DIM;
  const float* bg = g_  + layer * HID_DIM;
  const float* bb = be_ + layer * HID_DIM;
  const float* bm = m_  + layer * HID_DIM;
  const float* bv = v_  + layer * HID_DIM;

  // Stage 0: h0 = (1+eps)*x + agg  (16 rows x 128 cols, 16 floats/thread)
  {
    const int r  = tid >> 3;          // 0..15
    const int c0 = (tid & 7) * 16;    // 0,16,...,112
    const int node = rowBase + r;
    float* d = &h0[r * LSTR + c0];
    if (node < nNodes) {
      const float* xp = x   + (size_t)node * HID_DIM + c0;
      const float* ap = agg + (size_t)node * HID_DIM + c0;
#pragma unroll
      for (int j = 0; j < 16; ++j) d[j] = eps1 * xp[j] + ap[j];
    } else {
#pragma unroll
      for (int j = 0; j < 16; ++j) d[j] = 0.f;
    }
  }
  __syncthreads();

  const int wave = tid >> 5;      // 0..3
  const int lane = tid & 31;
  const int lrow = lane & 15;     // M index within tile (A rows / C cols)
  const int hi   = lane >> 4;     // lane half selects K pair / C row half
  const int ct0  = wave * 2;      // this wave's two column tiles
  const int ct1  = ct0 + 1;
  const int c0   = ct0 * 16 + lrow;
  const int c1   = ct1 * 16 + lrow;

  // ---- GEMM 1: h1 = relu(h0 @ W1 + b1) ----
  {
    v8f a0, a1;
    const float bias0 = b1[c0], bias1 = b1[c1];
#pragma unroll
    for (int r = 0; r < 8; ++r) { a0[r] = bias0; a1[r] = bias1; }
#pragma unroll
    for (int k = 0; k < HID_DIM; k += 4) {
      // A frag (16x4): VGPR0 = K0|K2, VGPR1 = K1|K3 across lane halves
      v2f af;
      af.x = h0[lrow * LSTR + k + 2 * hi];
      af.y = h0[lrow * LSTR + k + 2 * hi + 1];
      // B frag (4x16): row of K striped across lanes, mirrored lane halves
      v2f bf0, bf1;
      bf0.x = W1[(size_t)(k + 2 * hi)     * HID_DIM + c0];
      bf0.y = W1[(size_t)(k + 2 * hi + 1) * HID_DIM + c0];
      bf1.x = W1[(size_t)(k + 2 * hi)     * HID_DIM + c1];
      bf1.y = W1[(size_t)(k + 2 * hi + 1) * HID_DIM + c1];
      a0 = wmma_f32_k4(af, bf0, a0);
      a1 = wmma_f32_k4(af, bf1, a1);
    }
#pragma unroll
    for (int r = 0; r < 8; ++r) {
      const int row = r + hi * 8;   // C layout: VGPR r -> M=r | M=r+8
      h1[row * LSTR + c0] = fmaxf(a0[r], 0.f);
      h1[row * LSTR + c1] = fmaxf(a1[r], 0.f);
    }
  }
  __syncthreads();

  // ---- GEMM 2 + BN: out = g*(relu(h1@W2+b2) - m)*rsqrt(v+eps) + be ----
  {
    v8f a0, a1;
    const float bias0 = b2[c0], bias1 = b2[c1];
#pragma unroll
    for (int r = 0; r < 8; ++r) { a0[r] = bias0; a1[r] = bias1; }
#pragma unroll
    for (int k = 0; k < HID_DIM; k += 4) {
      v2f af;
      af.x = h1[lrow * LSTR + k + 2 * hi];
      af.y = h1[lrow * LSTR + k + 2 * hi + 1];
      v2f bf0, bf1;
      bf0.x = W2[(size_t)(k + 2 * hi)     * HID_DIM + c0];
      bf0.y = W2[(size_t)(k + 2 * hi + 1) * HID_DIM + c0];
      bf1.x = W2[(size_t)(k + 2 * hi)     * HID_DIM + c1];
      bf1.y = W2[(size_t)(k + 2 * hi + 1) * HID_DIM + c1];
      a0 = wmma_f32_k4(af, bf0, a0);
      a1 = wmma_f32_k4(af, bf1, a1);
    }
    const float s0 = bg[c0] * rsqrtf(bv[c0] + BN_EPS);
    const float s1 = bg[c1] * rsqrtf(bv[c1] + BN_EPS);
    const float m0 = bm[c0], m1 = bm[c1];
    const float o0 = bb[c0], o1 = bb[c1];
#pragma unroll
    for (int r = 0; r < 8; ++r) {
      const int row  = r + hi * 8;
      const int node = rowBase + row;
      if (node < nNodes) {
        out[(size_t)node * HID_DIM + c0] = s0 * (fmaxf(a0[r], 0.f) - m0) + o0;
        out[(size_t)node * HID_DIM + c1] = s1 * (fmaxf(a1[r], 0.f) - m1) + o1;
      }
    }
  }
}

// ---------------------------------------------------------------------------
// Per-graph mean pool accumulation: sums[batch[n]] += x[n], cnts[batch[n]] += 1
// ---------------------------------------------------------------------------
__global__ __launch_bounds__(256)
void pool_kernel(const float* __restrict__ x, const int* __restrict__ batch,
                 float* __restrict__ sums, float* __restrict__ cnts, int nNodes) {
  int tid = blockIdx.x * 256 + threadIdx.x;
  int n = tid >> 5;
  int q = tid & 31;
  if (n < nNodes) {
    int g = batch[n];
    const float4 v = *(const float4*)(x + (size_t)n * HID_DIM + q * 4);
    float* o = sums + (size_t)g * HID_DIM + q * 4;
    atomicAdd(o + 0, v.x);
    atomicAdd(o + 1, v.y);
    atomicAdd(o + 2, v.z);
    atomicAdd(o + 3, v.w);
    if (q == 0) atomicAdd(&cnts[g], 1.0f);
  }
}

// ---------------------------------------------------------------------------
// Head: pooled = sums/max(cnt,1); h = relu(pooled@lin1+b); logits = h@lin2+b;
// out = log_softmax(logits).  One block (128 thr) per graph — negligible cost.
// ---------------------------------------------------------------------------
__global__ __launch_bounds__(128)
void head_kernel(const float* __restrict__ sums, const float* __restrict__ cnts,
                 const float* __restrict__ l1w, const float* __restrict__ l1b,
                 const float* __restrict__ l2w, const float* __restrict__ l2b,
                 float* __restrict__ out, int outDim) {
  __shared__ float pooled[HID_DIM];
  __shared__ float hbuf[HID_DIM];
  __shared__ float logits[16];
  const int g = blockIdx.x;
  const int t = threadIdx.x;

  const float inv = 1.0f / fmaxf(cnts[g], 1.0f);
  pooled[t] = sums[(size_t)g * HID_DIM + t] * inv;
  __syncthreads();

  float acc = l1b[t];
  for (int k = 0; k < HID_DIM; ++k) acc += pooled[k] * l1w[(size_t)k * HID_DIM + t];
  hbuf[t] = fmaxf(acc, 0.f);
  __syncthreads();

  if (t < outDim) {
    float a = l2b[t];
    for (int k = 0; k < HID_DIM; ++k) a += hbuf[k] * l2w[(size_t)k * outDim + t];
    logits[t] = a;
  }
  __syncthreads();

  if (t == 0) {
    float mx = -1e30f;
    for (int j = 0; j < outDim; ++j) mx = fmaxf(mx, logits[j]);
    float se = 0.f;
    for (int j = 0; j < outDim; ++j) se += expf(logits[j] - mx);
    const float lse = mx + logf(se);
    for (int j = 0; j < outDim; ++j) out[(size_t)g * outDim + j] = logits[j] - lse;
  }
}

// ---------------------------------------------------------------------------
extern "C" void kernel_launch(void* const* d_in, const int* in_sizes, int n_in,
                              void* d_out, int out_size, void* d_ws, size_t ws_size,
                              hipStream_t stream) {
  (void)n_in; (void)ws_size;
  const float* x0   = (const float*)d_in[0];
  const int*   ei   = (const int*)d_in[1];
  const int*   bat  = (const int*)d_in[2];
  const float* W1   = (const float*)d_in[3];
  const float* b1   = (const float*)d_in[4];
  const float* W2   = (const float*)d_in[5];
  const float* b2   = (const float*)d_in[6];
  const float* bng  = (const float*)d_in[7];
  const float* bnb  = (const float*)d_in[8];
  const float* bnm  = (const float*)d_in[9];
  const float* bnv  = (const float*)d_in[10];
  const float* epsg = (const float*)d_in[11];
  const float* l1w  = (const float*)d_in[12];
  const float* l1b  = (const float*)d_in[13];
  const float* l2w  = (const float*)d_in[14];
  const float* l2b  = (const float*)d_in[15];

  const int nNodes  = in_sizes[0] / HID_DIM;  // IN_DIM == HID == 128
  const int nEdges  = in_sizes[1] / 2;
  const int outDim  = 10;
  const int nGraphs = out_size / outDim;

  const int* src = ei;
  const int* dst = ei + nEdges;

  const size_t featBytes = (size_t)nNodes * HID_DIM * sizeof(float);
  float* bufA = (float*)d_ws;
  float* bufB = (float*)((char*)d_ws + featBytes);
  float* sums = (float*)((char*)d_ws + 2 * featBytes);
  float* cnts = sums + (size_t)nGraphs * HID_DIM;
  float* bufs[2] = {bufA, bufB};

  const int scatterBlocks = (nEdges * 32 + 255) / 256;
  const int mlpBlocks     = (nNodes + 15) / 16;

  const float* cur = x0;
  for (int l = 0; l < 4; ++l) {
    float* agg = bufs[l & 1];
    hipMemsetAsync(agg, 0, featBytes, stream);
    scatter_add_kernel<<<scatterBlocks, 256, 0, stream>>>(cur, src, dst, agg, nEdges);
    // out aliases agg: each block reads its rows into LDS before overwriting.
    gin_mlp_kernel<<<mlpBlocks, 128, 0, stream>>>(
        cur, agg, epsg, l, W1, b1, W2, b2, bng, bnb, bnm, bnv, agg, nNodes);
    cur = agg;
  }

  hipMemsetAsync(sums, 0, ((size_t)nGraphs * HID_DIM + nGraphs) * sizeof(float), stream);
  pool_kernel<<<(nNodes * 32 + 255) / 256, 256, 0, stream>>>(cur, bat, sums, cnts, nNodes);
  head_kernel<<<nGraphs, 128, 0, stream>>>(sums, cnts, l1w, l1b, l2w, l2b,
                                           (float*)d_out, outDim);
}